// RollingSignatureFeatures_48473000902899
// MI455X (gfx1250) — compile-verified
//
#include <hip/hip_runtime.h>

// CDNA5 / gfx1250 WMMA vector types
typedef float v2f __attribute__((ext_vector_type(2)));
typedef float v8f __attribute__((ext_vector_type(8)));

#define B_DIM 64
#define S_DIM 2048
#define D_DIM 15
#define WAVES_PER_BLOCK 8   // 256 threads = 8 wave32

// One wave32 computes one (b, s) output row of 272 f32:
//   level1 (16) + level2 (16x16) via V_WMMA_F32_16X16X4_F32.
//
// Roofline: ~150 MB total traffic vs ~0.24 GFLOP -> purely HBM-store bound
// (~6.5 us floor at 23.3 TB/s). Input (7.9 MB) is fully L2-resident, so we
// spend the instruction budget on the WMMA outer-product-sum and NT stores.
__global__ __launch_bounds__(256)
void RollingSignatureFeatures_48473000902899_kernel(const float* __restrict__ feat,
                                                    float* __restrict__ out)
{
    const int wave = blockIdx.x * WAVES_PER_BLOCK + (threadIdx.x >> 5);
    const int lane = threadIdx.x & 31;
    const int m    = lane & 15;   // component index (0 = time, 1..15 = features)
    const int half = lane >> 4;   // 0: K=0,1   1: K=2,3
    const int k0   = half << 1;

    const int b = wave >> 11;          // S_DIM == 2048
    const int s = wave & (S_DIM - 1);

    const float inv_sm1 = 1.0f / (float)(S_DIM - 1);
    const bool  is_time = (m == 0);
    // In-bounds column for the load even when m==0 (value is replaced by time).
    const int   col     = is_time ? 0 : (m - 1);

    // Augmented window rows row[k0], row[k0+1], row[k0+2] at component m.
    // Window row w corresponds to t = clamp(s - 4 + w, 0).
    float r[3];
#pragma unroll
    for (int i = 0; i < 3; ++i) {
        int t = s - 4 + k0 + i;
        if (t < 0) t = 0;
        const float ld = feat[((size_t)b * S_DIM + t) * D_DIM + col];
        r[i] = is_time ? (float)t * inv_sm1 : ld;   // select, no divergent load
    }
    // Row 4 (t = s) for level1; all lanes fetch (uniform control flow, shared lines).
    const float ld4 = feat[((size_t)b * S_DIM + s) * D_DIM + col];
    const float r4  = is_time ? (float)s * inv_sm1 : ld4;

    // A[i, k] = increments[k][i] = row[k+1][i] - row[k][i]
    // B[k, j] = left[k][j]       = row[k][j]
    v2f a, bm;
    a.x  = r[1] - r[0];   // increments[k0]   at component m
    a.y  = r[2] - r[1];   // increments[k0+1] at component m
    bm.x = r[0];          // left[k0]         at component m
    bm.y = r[1];          // left[k0+1]       at component m

    v8f c = {0.f, 0.f, 0.f, 0.f, 0.f, 0.f, 0.f, 0.f};
    // D = A(16x4) * B(4x16) + 0, all f32 -> exact match to the reference einsum
    c = __builtin_amdgcn_wmma_f32_16x16x4_f32(
            /*neg_a=*/false, a, /*neg_b=*/false, bm,
            /*c_mod=*/(short)0, c, /*reuse_a=*/false, /*reuse_b=*/false);

    float* o = out + (size_t)wave * 272;   // wave == b*S + s

    // level1 = row4 - row0 ; lanes 0..15 hold row0 in r[0]
    if (lane < 16) {
        __builtin_nontemporal_store(r4 - r[0], o + m);
    }

    // level2: C/D layout -> VGPR rr, lane (half,m) holds element (M = rr + 8*half, N = m)
#pragma unroll
    for (int rr = 0; rr < 8; ++rr) {
        const int M = rr + (half << 3);
        __builtin_nontemporal_store(c[rr], o + 16 + M * 16 + m);
    }
}

extern "C" void kernel_launch(void* const* d_in, const int* in_sizes, int n_in,
                              void* d_out, int out_size, void* d_ws, size_t ws_size,
                              hipStream_t stream) {
    (void)in_sizes; (void)n_in; (void)d_ws; (void)ws_size; (void)out_size;
    const float* feat = (const float*)d_in[0];
    float* out = (float*)d_out;

    const int total_waves = B_DIM * S_DIM;                 // 131072
    const int blocks = total_waves / WAVES_PER_BLOCK;      // 16384
    RollingSignatureFeatures_48473000902899_kernel<<<blocks, WAVES_PER_BLOCK * 32, 0, stream>>>(feat, out);
}